// MeanToERA5_3530463117835
// MI455X (gfx1250) — compile-verified
//
#include <hip/hip_runtime.h>
#include <stdint.h>

// Problem constants (match reference)
#define NCLUST   4096
#define NPTS     (512 * 512)        // 262144 points
#define NBATCH   256                // 32*8 channels
#define CHG      8                  // channels per workgroup
#define NCHG     (NBATCH / CHG)     // 32 channel groups
#define NSLICE   8                  // spatial slices per channel group
#define SCAT_TPB 1024
#define SLICE_PTS (NPTS / NSLICE)   // 32768

typedef float v2f __attribute__((ext_vector_type(2)));
typedef float v8f __attribute__((ext_vector_type(8)));

// ---------------------------------------------------------------------------
// Kernel 1: per-cluster counts (float, to match reference segment_sum of ones)
// ---------------------------------------------------------------------------
__global__ __launch_bounds__(256)
void mean2era5_count(const int* __restrict__ mapping, float* __restrict__ counts) {
    int i = blockIdx.x * blockDim.x + threadIdx.x;
    if (i < NPTS) atomicAdd(&counts[mapping[i]], 1.0f);
}

// ---------------------------------------------------------------------------
// Kernel 2: scatter-sum. Each workgroup owns CHG=8 channels and one spatial
// slice; accumulates into a 128 KB LDS histogram (channel-major so LDS bank
// = cluster mod 64 -> good spread for random clusters), then flushes with
// global float atomics into sums[b*NCLUST + c] (accumulated in d_out).
// ---------------------------------------------------------------------------
__global__ __launch_bounds__(SCAT_TPB)
void mean2era5_scatter(const float* __restrict__ vals,
                       const int* __restrict__ mapping,
                       float* __restrict__ sums) {
    __shared__ float lsum[CHG][NCLUST];              // 8*4096*4 = 128 KB LDS
    const int wg    = blockIdx.x;                    // [0, NCHG*NSLICE)
    const int chg   = wg / NSLICE;
    const int slice = wg % NSLICE;
    const int b0    = chg * CHG;

    float* lflat = &lsum[0][0];
    for (int i = threadIdx.x; i < CHG * NCLUST; i += SCAT_TPB) lflat[i] = 0.0f;
    __syncthreads();

    const int base = slice * SLICE_PTS;
    for (int off = threadIdx.x; off < SLICE_PTS; off += SCAT_TPB) {
        const int n = base + off;
        const int c = mapping[n];
        // prefetch next iteration's first value stream (speculative, OOB-safe)
        __builtin_prefetch(&vals[(size_t)b0 * NPTS + n + SCAT_TPB], 0, 1);
#pragma unroll
        for (int ch = 0; ch < CHG; ++ch) {
            const float v = vals[(size_t)(b0 + ch) * NPTS + n];
            atomicAdd(&lsum[ch][c], v);              // ds_add_f32
        }
    }
    __syncthreads();

    for (int i = threadIdx.x; i < CHG * NCLUST; i += SCAT_TPB) {
        const int ch = i / NCLUST;
        const int c  = i % NCLUST;
        atomicAdd(&sums[(size_t)(b0 + ch) * NCLUST + c], lflat[i]);
    }
}

// ---------------------------------------------------------------------------
// Kernel 3: normalize with V_WMMA_F32_16X16X4_F32.
// Per wave: one 16x16 tile of means (16 batch rows x 16 cluster cols).
// D = sum_{j=0..3} A_j x B_j where A_j[m][k] = sums[m0+m][c0+4j+k] and
// B_j[k][n] = inv[c0+n] iff n == 4j+k else 0  -> D[m][n] = sums*inv exactly
// (every element gets exactly one nonzero product; zero adds are exact).
// VGPR layouts per ISA 7.12.2:
//   A 16x4 : lane<16 -> M=lane,    v0=K0,v1=K1 ; lane>=16 -> M=lane-16, K2,K3
//   B 4x16 : lane<16 -> N=lane,    v0=K0,v1=K1 ; lane>=16 -> N=lane-16, K2,K3
//   C/D    : vgpr v  -> M = v + 8*(lane>=16), N = lane&15
// Exactly 4096 waves for 4096 tiles -> no partial waves, EXEC all ones.
// ---------------------------------------------------------------------------
__global__ __launch_bounds__(256)
void mean2era5_normalize(float* __restrict__ inout,        // sums in, means out
                         const float* __restrict__ counts) {
    const int wave = (blockIdx.x * blockDim.x + threadIdx.x) >> 5;  // [0,4096)
    const int lane = threadIdx.x & 31;
    const int half = lane >> 4;        // 0: lanes 0-15, 1: lanes 16-31
    const int lp   = lane & 15;

    const int m0 = (wave / (NCLUST / 16)) * 16;   // batch-row tile origin
    const int c0 = (wave % (NCLUST / 16)) * 16;   // cluster-col tile origin

    const float invv = 1.0f / counts[c0 + lp];    // reciprocal for column N=lp
    const int   k0   = 2 * half;                  // this half-wave holds K = k0, k0+1

    v8f acc = {};
#pragma unroll
    for (int j = 0; j < 4; ++j) {
        v2f a, b;
        const size_t row = (size_t)(m0 + lp) * NCLUST + c0 + 4 * j + k0;
        a.x = inout[row];
        a.y = inout[row + 1];
        b.x = (lp == 4 * j + k0)     ? invv : 0.0f;
        b.y = (lp == 4 * j + k0 + 1) ? invv : 0.0f;
        acc = __builtin_amdgcn_wmma_f32_16x16x4_f32(
            /*neg_a=*/false, a, /*neg_b=*/false, b,
            /*c_mod=*/(short)0, acc, /*reuse_a=*/false, /*reuse_b=*/false);
    }

#pragma unroll
    for (int v = 0; v < 8; ++v) {
        inout[(size_t)(m0 + v + 8 * half) * NCLUST + c0 + lp] = acc[v];
    }
}

// ---------------------------------------------------------------------------
// Host launcher
// ---------------------------------------------------------------------------
extern "C" void kernel_launch(void* const* d_in, const int* in_sizes, int n_in,
                              void* d_out, int out_size, void* d_ws, size_t ws_size,
                              hipStream_t stream) {
    const float* vals    = (const float*)d_in[0];   // [256, 262144] f32
    const int*   mapping = (const int*)d_in[1];     // [262144] i32
    float*       out     = (float*)d_out;           // [256, 4096] f32 (sums, then means)
    float*       counts  = (float*)d_ws;            // [4096] f32 scratch

    // zero accumulators (graph-capture safe async memsets)
    hipMemsetAsync(out, 0, (size_t)NBATCH * NCLUST * sizeof(float), stream);
    hipMemsetAsync(counts, 0, (size_t)NCLUST * sizeof(float), stream);

    mean2era5_count<<<NPTS / 256, 256, 0, stream>>>(mapping, counts);
    mean2era5_scatter<<<NCHG * NSLICE, SCAT_TPB, 0, stream>>>(vals, mapping, out);
    // 4096 tiles, 1 wave each -> 131072 threads -> 512 blocks of 256
    mean2era5_normalize<<<512, 256, 0, stream>>>(out, counts);
}